// MAMBAModel_26036091748685
// MI455X (gfx1250) — compile-verified
//
#include <hip/hip_runtime.h>
#include <hip/hip_bf16.h>
#include <math.h>

// ---------------- model dims (match reference) ----------------
#define B_    2
#define L_    1024
#define D_    1024
#define NL_   4
#define V_    50280
#define DI_   2048          // 2*D
#define DS_   16
#define DC_   4
#define DTR_  64            // D/16
#define M_    (B_ * L_)     // 2048 rows
#define EPS_  1e-5f
#define XPN_  (DTR_ + 2 * DS_)   // 96

typedef __attribute__((ext_vector_type(2))) float v2f;
typedef __attribute__((ext_vector_type(8))) float v8f;

// =====================================================================
// Generic fp32 WMMA GEMM:  C[m,n] = act( sum_k A[m,k]*W[n,k] + bias[n] )
//   A: M x K, row stride lda   (M multiple of 16; K multiple of 16)
//   W: N x K, row stride K     (weights row-major; plays the role of B^T)
//   C: M x N, row stride N
//   act: 0 = none, 1 = softplus(x + bias)
//
// One wave32 computes a 16x64 output tile with V_WMMA_F32_16X16X4_F32.
// K-loop steps by 16 (one 64B line per row): 20 batched b64 loads feed a
// burst of 16 WMMAs, letting the scheduler use partial s_wait_loadcnt
// instead of a full wait per load.
//
// Ragged-N handling: out-of-range columns clamp their weight pointer to
// row 0. Since D[:,n] depends only on B[:,n], which lives in lanes n and
// n+16 -- the very lanes whose stores are masked -- the garbage never
// reaches memory, so no input masking (cndmask) is needed at all.
// =====================================================================
__global__ __launch_bounds__(256) void gemm_f32_wmma(
    const float* __restrict__ A, int lda,
    const float* __restrict__ W,
    float* __restrict__ C,
    int M, int N, int K,
    const float* __restrict__ bias, int act)
{
    const int tilesN4 = (N + 63) >> 6;                  // 64-wide N groups
    const int wave    = threadIdx.x >> 5;
    const int lane    = threadIdx.x & 31;
    const int gtile   = blockIdx.x * 8 + wave;          // 8 waves / block
    const int tileM   = gtile / tilesN4;
    const int tileN4  = gtile - tileM * tilesN4;
    if (tileM * 16 >= M) return;                        // wave-uniform exit

    const int half   = lane >> 4;                       // 0: K pair {k,k+1}, 1: {k+2,k+3}
    const int lm     = lane & 15;                       // row (A) / column (B,C) index
    const int kshift = half * 2;

    const int mb = tileM * 16;
    const int nb = tileN4 * 64;

    const float* Arow = A + (size_t)(mb + lm) * lda + kshift;

    const float* Wrow[4];
    bool nvalid[4];
#pragma unroll
    for (int j = 0; j < 4; ++j) {
        const int nc = nb + j * 16 + lm;
        nvalid[j] = (nc < N);
        Wrow[j] = W + (size_t)(nvalid[j] ? nc : 0) * K + kshift;  // clamp, see header
    }

    v8f acc[4];
#pragma unroll
    for (int j = 0; j < 4; ++j) acc[j] = (v8f){0.f,0.f,0.f,0.f,0.f,0.f,0.f,0.f};

    for (int k = 0; k < K; k += 16) {
        // one prefetch per stream per 16-WMMA iteration, 2 lines ahead
        __builtin_prefetch((const void*)(Arow + k + 32), 0, 3);
#pragma unroll
        for (int j = 0; j < 4; ++j)
            __builtin_prefetch((const void*)(Wrow[j] + k + 32), 0, 3);

        // ---- batch all fragment loads (4 A + 16 B, distinct regs) ----
        v2f a[4], b[4][4];
#pragma unroll
        for (int u = 0; u < 4; ++u)
            a[u] = *(const v2f*)(Arow + k + 4 * u);
#pragma unroll
        for (int j = 0; j < 4; ++j)
#pragma unroll
            for (int u = 0; u < 4; ++u)
                b[j][u] = *(const v2f*)(Wrow[j] + k + 4 * u);

        // ---- 16 back-to-back WMMAs ----
#pragma unroll
        for (int u = 0; u < 4; ++u)
#pragma unroll
            for (int j = 0; j < 4; ++j)
                acc[j] = __builtin_amdgcn_wmma_f32_16x16x4_f32(
                    /*neg_a=*/false, a[u], /*neg_b=*/false, b[j][u],
                    /*c_mod=*/(short)0, acc[j], /*reuse_a=*/false, /*reuse_b=*/false);
    }

#pragma unroll
    for (int j = 0; j < 4; ++j) {
        if (!nvalid[j]) continue;
        const int ncol = nb + j * 16 + lm;
        float* Cp = C + (size_t)(mb + 8 * half) * N + ncol;
        const float bval = (act == 1) ? bias[ncol] : 0.f;
#pragma unroll
        for (int r = 0; r < 8; ++r) {                   // VGPR r -> row mb + 8*half + r
            float v = acc[j][r];
            if (act == 1) {                             // softplus(v + bias)
                v += bval;
                v = (v > 20.f) ? v : log1pf(expf(v));
            }
            Cp[(size_t)r * N] = v;
        }
    }
}

// =====================================================================
// Embedding gather: hidden[row,:] = emb[ids[row],:]   (256 thr x float4)
// =====================================================================
__global__ __launch_bounds__(256) void embed_kernel(
    const int* __restrict__ ids, const float* __restrict__ emb,
    float* __restrict__ hid)
{
    const int row = blockIdx.x;
    const int tok = ids[row];
    ((float4*)(hid + (size_t)row * D_))[threadIdx.x] =
        ((const float4*)(emb + (size_t)tok * D_))[threadIdx.x];
}

// =====================================================================
// residual (+)= hidden ; out = rmsnorm(residual) * w
// one block (256 threads) per row, D=1024 -> float4 per thread
// =====================================================================
__global__ __launch_bounds__(256) void rmsnorm_kernel(
    const float* __restrict__ hidden, float* __restrict__ residual,
    const float* __restrict__ w, float* __restrict__ out, int first)
{
    const int row = blockIdx.x;
    const int tid = threadIdx.x;

    float4 hv = ((const float4*)(hidden + (size_t)row * D_))[tid];
    float4 rv;
    if (first) {
        rv = hv;
    } else {
        rv = ((const float4*)(residual + (size_t)row * D_))[tid];
        rv.x += hv.x; rv.y += hv.y; rv.z += hv.z; rv.w += hv.w;
    }
    ((float4*)(residual + (size_t)row * D_))[tid] = rv;

    float ss = rv.x * rv.x + rv.y * rv.y + rv.z * rv.z + rv.w * rv.w;
    for (int off = 16; off > 0; off >>= 1)
        ss += __shfl_down(ss, off, 32);                 // wave32 reduce

    __shared__ float red[8];
    const int wv = tid >> 5, ln = tid & 31;
    if (ln == 0) red[wv] = ss;
    __syncthreads();
    if (tid == 0) {
        float t = 0.f;
        for (int i = 0; i < 8; ++i) t += red[i];
        red[0] = rsqrtf(t * (1.0f / D_) + EPS_);
    }
    __syncthreads();
    const float scale = red[0];

    float4 wvv = ((const float4*)w)[tid];
    float4 o;
    o.x = rv.x * scale * wvv.x;
    o.y = rv.y * scale * wvv.y;
    o.z = rv.z * scale * wvv.z;
    o.w = rv.w * scale * wvv.w;
    ((float4*)(out + (size_t)row * D_))[tid] = o;
}

// =====================================================================
// depthwise causal conv (DC=4) + bias + silu over the x half of xz
// one thread per (b,l,d)
// =====================================================================
__global__ __launch_bounds__(256) void conv_silu_kernel(
    const float* __restrict__ xz, const float* __restrict__ cw,
    const float* __restrict__ cb, float* __restrict__ x)
{
    const int idx = blockIdx.x * blockDim.x + threadIdx.x;
    if (idx >= M_ * DI_) return;
    const int d = idx % DI_;
    const int n = idx / DI_;            // b*L + l
    const int l = n % L_;

    float acc = cb[d];
#pragma unroll
    for (int j = 0; j < DC_; ++j) {
        const int ls = l - (DC_ - 1) + j;
        if (ls >= 0)
            acc += xz[(size_t)(n - l + ls) * (2 * DI_) + d] * cw[d * DC_ + j];
    }
    x[idx] = acc / (1.f + expf(-acc));  // silu
}

// =====================================================================
// Selective scan: sequential over L, one thread per (b,d) channel,
// 16 states in registers, exp on the TRANS pipe.
// y = (scan_out + x*D_skip) * silu(z)
// =====================================================================
__global__ __launch_bounds__(256) void scan_kernel(
    const float* __restrict__ dt,   const float* __restrict__ dbl,
    const float* __restrict__ x,    const float* __restrict__ xz,
    const float* __restrict__ A_log,const float* __restrict__ D_skip,
    float* __restrict__ y)
{
    const int idx = blockIdx.x * blockDim.x + threadIdx.x;
    if (idx >= B_ * DI_) return;
    const int d = idx % DI_;
    const int b = idx / DI_;

    float Ar[DS_], h[DS_];
#pragma unroll
    for (int s = 0; s < DS_; ++s) { Ar[s] = -expf(A_log[d * DS_ + s]); h[s] = 0.f; }
    const float dsk = D_skip[d];

    for (int l = 0; l < L_; ++l) {
        const int n = b * L_ + l;
        const float dtv = dt[(size_t)n * DI_ + d];
        const float xv  = x [(size_t)n * DI_ + d];
        const float zv  = xz[(size_t)n * (2 * DI_) + DI_ + d];
        const float* bc = dbl + (size_t)n * XPN_;       // [0:64)=dt_raw [64:80)=B [80:96)=C
        float yv = 0.f;
#pragma unroll
        for (int s = 0; s < DS_; ++s) {
            const float dA = expf(dtv * Ar[s]);
            h[s] = dA * h[s] + dtv * bc[DTR_ + s] * xv;
            yv  += h[s] * bc[DTR_ + DS_ + s];
        }
        yv += xv * dsk;
        const float sz = zv / (1.f + expf(-zv));        // silu(z)
        y[(size_t)n * DI_ + d] = yv * sz;
    }
}

// =====================================================================
extern "C" void kernel_launch(void* const* d_in, const int* in_sizes, int n_in,
                              void* d_out, int out_size, void* d_ws, size_t ws_size,
                              hipStream_t stream) {
    const int*   ids        = (const int*)  d_in[0];
    const float* emb        = (const float*)d_in[1];
    const float* norm_w     = (const float*)d_in[2];
    const float* in_proj_w  = (const float*)d_in[3];
    const float* conv_w     = (const float*)d_in[4];
    const float* conv_b     = (const float*)d_in[5];
    const float* x_proj_w   = (const float*)d_in[6];
    const float* dt_proj_w  = (const float*)d_in[7];
    const float* dt_proj_b  = (const float*)d_in[8];
    const float* A_log      = (const float*)d_in[9];
    const float* D_skip     = (const float*)d_in[10];
    const float* out_proj_w = (const float*)d_in[11];
    const float* norm_f_w   = (const float*)d_in[12];
    float*       logits     = (float*)d_out;

    // ---- workspace layout (floats), ~110 MB total ----
    float* ws       = (float*)d_ws;
    float* residual = ws;  ws += (size_t)M_ * D_;
    float* hidden   = ws;  ws += (size_t)M_ * D_;
    float* hbuf     = ws;  ws += (size_t)M_ * D_;
    float* xz       = ws;  ws += (size_t)M_ * 2 * DI_;
    float* xbuf     = ws;  ws += (size_t)M_ * DI_;
    float* dbl      = ws;  ws += (size_t)M_ * XPN_;
    float* dtbuf    = ws;  ws += (size_t)M_ * DI_;
    float* ybuf     = ws;  ws += (size_t)M_ * DI_;

    auto gemm = [&](const float* A, int lda, const float* W, float* C,
                    int M, int N, int K, const float* bias, int act) {
        const int tiles = ((M + 15) / 16) * ((N + 63) / 64);   // 16x64 per wave
        gemm_f32_wmma<<<(tiles + 7) / 8, 256, 0, stream>>>(A, lda, W, C, M, N, K, bias, act);
    };

    embed_kernel<<<M_, 256, 0, stream>>>(ids, emb, hidden);

    for (int i = 0; i < NL_; ++i) {
        // residual (+)= hidden ; hbuf = rmsnorm(residual) * norm_w[i]
        rmsnorm_kernel<<<M_, 256, 0, stream>>>(hidden, residual, norm_w + (size_t)i * D_,
                                               hbuf, i == 0);
        // xz = hbuf @ in_proj_w[i]^T           (2048 x 4096 x 1024)
        gemm(hbuf, D_, in_proj_w + (size_t)i * 2 * DI_ * D_, xz, M_, 2 * DI_, D_, nullptr, 0);
        // x = silu(causal_conv(xz[:, :DI]) + cb)
        conv_silu_kernel<<<(M_ * DI_ + 255) / 256, 256, 0, stream>>>(
            xz, conv_w + (size_t)i * DI_ * DC_, conv_b + (size_t)i * DI_, xbuf);
        // dbl = x @ x_proj_w[i]^T              (2048 x 96 x 2048)
        gemm(xbuf, DI_, x_proj_w + (size_t)i * XPN_ * DI_, dbl, M_, XPN_, DI_, nullptr, 0);
        // dt = softplus(dbl[:, :64] @ dt_proj_w[i]^T + dtb)   (2048 x 2048 x 64, lda=96)
        gemm(dbl, XPN_, dt_proj_w + (size_t)i * DI_ * DTR_, dtbuf, M_, DI_, DTR_,
             dt_proj_b + (size_t)i * DI_, 1);
        // selective scan (+ D skip + silu(z) gate)
        scan_kernel<<<(B_ * DI_ + 255) / 256, 256, 0, stream>>>(
            dtbuf, dbl, xbuf, xz, A_log + (size_t)i * DI_ * DS_,
            D_skip + (size_t)i * DI_, ybuf);
        // hidden = y @ out_proj_w[i]^T         (2048 x 1024 x 2048)
        gemm(ybuf, DI_, out_proj_w + (size_t)i * D_ * DI_, hidden, M_, D_, DI_, nullptr, 0);
    }

    // final residual + rmsnorm, then logits = hbuf @ emb^T  (2048 x 50280 x 1024)
    rmsnorm_kernel<<<M_, 256, 0, stream>>>(hidden, residual, norm_f_w, hbuf, 0);
    gemm(hbuf, D_, emb, logits, M_, V_, D_, nullptr, 0);
}